// PoolingPMATopK_31645319037440
// MI455X (gfx1250) — compile-verified
//
#include <hip/hip_runtime.h>

typedef __attribute__((ext_vector_type(2))) float v2f;
typedef __attribute__((ext_vector_type(8))) float v8f;

#define B_    16
#define N_    8192
#define H_    768
#define Q_    32
#define K_    128
#define SCALE 0.28867513459481287f   /* 12^-0.5 */

// ---------------------------------------------------------------------------
// Phase 1: scores[b,q,n] = sum_h seed[q,h] * input[b,n,h]
// One workgroup = 1 batch x 128 columns of N.  8 waves; each wave owns a
// 16x32 output strip = TWO 16x16 tiles sharing one A (seed) operand, so the
// K-loop issues 1 A-load + 2 B-loads per 2 WMMAs (A reuse, two independent
// accumulation chains).  K-loop: 192 x 2 x v_wmma_f32_16x16x4_f32.
//
// fp32 16x4 A layout (ISA 7.12.2): lanes 0-15 hold row M=lane, K={0,1} in the
// two VGPRs; lanes 16-31 the same rows with K={2,3}.  B (4x16) mirrors with N
// striped across lanes -> contiguous float2 per lane from row-major memory.
// ---------------------------------------------------------------------------
__global__ __launch_bounds__(256)
void pma_scores_wmma(const float* __restrict__ input,
                     const float* __restrict__ seed,
                     float* __restrict__ scores) {
  const int blk    = blockIdx.x;
  const int b      = blk >> 6;         // 64 chunks of 128 columns
  const int nchunk = blk & 63;
  const int tid    = threadIdx.x;
  const int wave   = tid >> 5;
  const int lane   = tid & 31;
  const int m_tile = wave >> 2;        // 0..1  (q rows 0-15 / 16-31)
  const int n_pair = wave & 3;         // 0..3  (pair of 16-wide n tiles)
  const int n_base = nchunk * 128 + n_pair * 32;
  const int lane_lo = lane & 15;
  const int lane_hi = lane >> 4;       // 0 -> K 0,1 ; 1 -> K 2,3

  const float* arow  = seed  + (size_t)(m_tile * 16 + lane_lo) * H_ + lane_hi * 2;
  const float* brow0 = input + ((size_t)b * N_ + (n_base + lane_lo)) * (size_t)H_
                             + lane_hi * 2;
  const float* brow1 = brow0 + (size_t)16 * H_;

  v8f acc0 = {};
  v8f acc1 = {};
#pragma unroll 4
  for (int h = 0; h < H_; h += 4) {
    v2f a  = *(const v2f*)(arow  + h);
    v2f b0 = *(const v2f*)(brow0 + h);
    v2f b1 = *(const v2f*)(brow1 + h);
    acc0 = __builtin_amdgcn_wmma_f32_16x16x4_f32(false, a, false, b0,
                                                 (short)0, acc0, false, false);
    acc1 = __builtin_amdgcn_wmma_f32_16x16x4_f32(false, a, false, b1,
                                                 (short)0, acc1, false, false);
  }

  // C/D layout: VGPR i -> lanes 0-15: (M=i, N=lane); lanes 16-31: (M=i+8, N=lane-16)
  float* srow0 = scores + ((size_t)b * Q_ + m_tile * 16 + lane_hi * 8) * (size_t)N_
                        + (n_base + lane_lo);
  float* srow1 = srow0 + 16;
#pragma unroll
  for (int i = 0; i < 8; ++i) {
    srow0[(size_t)i * N_] = acc0[i];
    srow1[(size_t)i * N_] = acc1[i];
  }
}

// ---------------------------------------------------------------------------
// Phase 2: per (b,q): top-128 of 8192 scores, softmax, weighted gather of V.
// wave32 shuffle reduction (5 lane-permute steps) + tiny cross-wave LDS scan:
// 2 barriers per selection round instead of ~10.
// ---------------------------------------------------------------------------
__device__ __forceinline__ unsigned long long packKey(float f, int idx) {
  unsigned u = __float_as_uint(f);
  u = (u & 0x80000000u) ? ~u : (u | 0x80000000u);     // monotonic total order
  return ((unsigned long long)u << 32) | (unsigned)(0xFFFFFFFFu - (unsigned)idx);
}

__device__ __forceinline__ unsigned long long localMax32(const float* v, int base) {
  unsigned long long m = 0ull;
#pragma unroll
  for (int i = 0; i < 32; ++i) {
    unsigned long long k = packKey(v[i], base + i);
    if (k > m) m = k;
  }
  return m;
}

__global__ __launch_bounds__(256)
void pma_topk_softmax_gather(const float* __restrict__ input,
                             const float* __restrict__ scores,
                             float* __restrict__ out) {
  __shared__ unsigned long long wmax[8];   // per-wave maxima
  __shared__ float topVal[K_];
  __shared__ int   topIdx[K_];
  __shared__ float attn[K_];
  __shared__ float ssum;

  const int bq   = blockIdx.x;          // b*32 + q
  const int b    = bq >> 5;
  const int tid  = threadIdx.x;
  const int wave = tid >> 5;
  const int lane = tid & 31;

  // Each thread owns 32 contiguous scores in registers.
  const float* srow = scores + (size_t)bq * N_;
  const int base = tid * 32;
  float v[32];
#pragma unroll
  for (int i = 0; i < 32; ++i) v[i] = srow[base + i];

  unsigned long long myKey = localMax32(v, base);

  // 128 rounds of argmax-and-remove (ties -> lowest index, like lax.top_k).
  for (int j = 0; j < K_; ++j) {
    // wave32 max-reduce via lane shuffles (no barriers)
    unsigned long long k = myKey;
#pragma unroll
    for (int d = 16; d >= 1; d >>= 1) {
      unsigned long long o = __shfl_xor(k, d, 32);
      if (o > k) k = o;
    }
    if (lane == 0) wmax[wave] = k;
    __syncthreads();

    // every thread redundantly scans the 8 wave maxima (no second barrier)
    unsigned long long w = wmax[0];
#pragma unroll
    for (int i = 1; i < 8; ++i)
      if (wmax[i] > w) w = wmax[i];

    const int widx = (int)(0xFFFFFFFFu - (unsigned)(w & 0xFFFFFFFFull));
    if (tid == 0) {
      unsigned u = (unsigned)(w >> 32);
      u = (u & 0x80000000u) ? (u & 0x7FFFFFFFu) : ~u;
      topVal[j] = __uint_as_float(u);
      topIdx[j] = widx;
    }
    if ((widx >> 5) == tid) {            // owner removes and rescans its 32
      v[widx & 31] = -__builtin_inff();
      myKey = localMax32(v, base);
    }
    __syncthreads();                     // wmax reads done before next round's writes
  }

  // Softmax over the 128 selected values (scaled by n_heads^-1/2).
  if (tid < K_) attn[tid] = __expf((topVal[tid] - topVal[0]) * SCALE);
  __syncthreads();
  if (tid == 0) {
    float s = 0.f;
#pragma unroll
    for (int j = 0; j < K_; ++j) s += attn[j];
    ssum = s;
  }
  __syncthreads();
  const float inv = 1.0f / ssum;

  // Weighted gather: out[b,q,h] = sum_j attn[j] * input[b, idx[j], h].
  const float* vb = input + (size_t)b * N_ * H_;
  float* orow = out + (size_t)bq * H_;
  for (int h = tid; h < H_; h += 256) {
    float acc = 0.f;
#pragma unroll 8
    for (int j = 0; j < K_; ++j)
      acc += attn[j] * vb[(size_t)topIdx[j] * H_ + h];
    orow[h] = acc * inv;
  }
}

extern "C" void kernel_launch(void* const* d_in, const int* in_sizes, int n_in,
                              void* d_out, int out_size, void* d_ws, size_t ws_size,
                              hipStream_t stream) {
  const float* input = (const float*)d_in[0];   // [16, 8192, 768] fp32
  const float* seed  = (const float*)d_in[1];   // [1, 32, 768]    fp32
  float* out    = (float*)d_out;                // [16, 32, 768]   fp32
  float* scores = (float*)d_ws;                 // [16, 32, 8192]  fp32 = 16 MB

  pma_scores_wmma<<<dim3(B_ * (N_ / 128)), dim3(256), 0, stream>>>(input, seed, scores);
  pma_topk_softmax_gather<<<dim3(B_ * Q_), dim3(256), 0, stream>>>(input, scores, out);
}